// ConvCaps_12137577578718
// MI455X (gfx1250) — compile-verified
//
#include <hip/hip_runtime.h>
#include <hip/hip_bf16.h>

// ---------------------------------------------------------------------------
// EM-routing ConvCaps for MI455X (gfx1250, wave32).
//   Kernel 1: vote GEMMs via V_WMMA_F32_16X16X4_F32 (K=4 == pose dim, exact fit)
//   Kernel 2: fused EM routing, one workgroup per output position
// ---------------------------------------------------------------------------

typedef __attribute__((ext_vector_type(2))) float v2f;
typedef __attribute__((ext_vector_type(8))) float v8f;

#define BATCH   8
#define HH      14
#define OHW     6
#define NPOS    288          // 8*6*6 output positions
#define NCAP    288          // K*K*B_IN input capsules per position
#define CC      32           // output capsules
#define PS      16           // pose size (4x4)
#define CPAD    17           // bank-conflict-free stride for [c][p] arrays
#define RPAD    33           // bank-conflict-free stride for r[n][c]

// ---------------------------------------------------------------------------
// Kernel 1: votes. One wave per (n, pos_tile-of-4); loops 8 c-tiles.
//   A (16x4) = 4 positions x 4 pose rows, K = j
//   B (4x16) = W[n] columns: 4 capsules x 4 l, K = j
//   D (16x16)= v[pos, n, c, i, l]
// v layout in workspace: v[pos][n][c][i*4+l]  (p contiguous for kernel 2)
// ---------------------------------------------------------------------------
__global__ __launch_bounds__(256) void convcaps_votes_wmma(
    const float* __restrict__ x,     // (8,14,14,32,16)
    const float* __restrict__ w,     // (288,32,4,4)
    float* __restrict__ v)           // (288 pos, 288 n, 32 c, 16 p)
{
    const int wid  = (blockIdx.x * blockDim.x + threadIdx.x) >> 5;
    const int lane = threadIdx.x & 31;
    const int n  = wid % NCAP;
    const int pt = wid / NCAP;              // 0..71 position tiles
    if (pt >= NPOS / 4) return;             // wave-uniform guard

    const int half = lane >> 4;             // selects K pair {0,1} vs {2,3}
    const int m    = lane & 15;

    // ---- A operand: pose[pos, n, i, j] ------------------------------------
    const int pos = pt * 4 + (m >> 2);
    const int i   = m & 3;
    const int b   = pos / 36;
    const int oh  = (pos / OHW) % OHW;
    const int ow  = pos % OHW;
    const int kh  = n / 96;                 // n = (kh*3+kw)*32 + bi
    const int kw  = (n / 32) % 3;
    const int bi  = n & 31;
    const long xoff =
        ((((long)(b * HH + oh * 2 + kh)) * HH + (ow * 2 + kw)) * 32 + bi) * PS
        + i * 4 + half * 2;                 // j = half*2, half*2+1 (contig)
    v2f A;
    A.x = x[xoff];
    A.y = x[xoff + 1];

    // ---- column decode for B and D ----------------------------------------
    const int cl = (lane & 15) >> 2;        // capsule within c-tile
    const int l  = lane & 3;                // pose column
    const int j0 = half * 2;

    #pragma unroll
    for (int ct = 0; ct < 8; ++ct) {
        const int c = ct * 4 + cl;
        const long woff = (((long)n * CC + c) * 4 + j0) * 4 + l; // W[n][c][j][l]
        v2f B;
        B.x = w[woff];
        B.y = w[woff + 4];                  // j0+1 row, stride 4 floats

        v8f D = {};
        D = __builtin_amdgcn_wmma_f32_16x16x4_f32(
                /*neg_a=*/false, A, /*neg_b=*/false, B,
                /*c_mod=*/(short)0, D, /*reuse_a=*/false, /*reuse_b=*/false);

        const int Ncol = lane & 15;
        #pragma unroll
        for (int r = 0; r < 8; ++r) {
            const int M    = r + half * 8;  // C/D layout: lanes>=16 hold M=8..15
            const int posl = M >> 2;
            const int ii   = M & 3;
            const int cc   = ct * 4 + (Ncol >> 2);
            const int ll   = Ncol & 3;
            const long off =
                ((((long)(pt * 4 + posl)) * NCAP + n) * CC + cc) * PS + ii * 4 + ll;
            v[off] = D[r];
        }
    }
}

// ---------------------------------------------------------------------------
// Kernel 2: EM routing. One 256-thread workgroup per output position.
// Lane <-> output capsule c (32 lanes == 32 capsules); 8 waves split n.
// ---------------------------------------------------------------------------
__global__ __launch_bounds__(256) void convcaps_em_routing(
    const float* __restrict__ v,       // votes (from kernel 1)
    const float* __restrict__ a,       // (8,14,14,32,1)
    const float* __restrict__ beta_u,  // (32,1)
    const float* __restrict__ beta_a,  // (32,1)
    float* __restrict__ out)           // p_out (147456) ++ a_out (9216)
{
    __shared__ float s_r[NCAP * RPAD];       // ln_ap -> r  (38 KB)
    __shared__ float s_mu[CC * CPAD];
    __shared__ float s_sig[CC * CPAD];
    __shared__ float s_hinv[CC * CPAD];      // 0.5/sigma
    __shared__ float s_amu[CC * CPAD];       // Sum coeff*v
    __shared__ float s_am2[CC * CPAD];       // Sum coeff*v^2
    __shared__ float s_rsum[CC];
    __shared__ float s_aout[CC];
    __shared__ float s_cst[CC];              // log(a) - 8*ln2pi - 0.5*Sum log sig
    __shared__ float s_ain[NCAP];

    const int tid  = threadIdx.x;
    const int lane = tid & 31;               // == capsule c for v passes
    const int wv   = tid >> 5;               // wave 0..7
    const int pos  = blockIdx.x;
    const int b    = pos / 36;
    const int oh   = (pos / OHW) % OHW;
    const int ow   = pos % OHW;

    const float eps   = 1e-6f;
    const float lam   = 1e-3f;
    const float LN2PI = 1.8378770664093453f;

    for (int n = tid; n < NCAP; n += 256) {
        const int kh = n / 96, kw = (n / 32) % 3, bi = n & 31;
        s_ain[n] = a[((b * HH + oh * 2 + kh) * HH + (ow * 2 + kw)) * 32 + bi];
    }
    __syncthreads();

    const float* vpos = v + (long)pos * NCAP * CC * PS;

    for (int it = 0; it < 3; ++it) {
        // ---------------- E step: r[n][c] ----------------------------------
        if (it == 0) {
            for (int idx = tid; idx < NCAP * CC; idx += 256) {
                const int n = idx >> 5, c = idx & 31;
                s_r[n * RPAD + c] = s_ain[n] * (1.0f / 32.0f);
            }
        } else {
            for (int n = wv; n < NCAP; n += 8) {
                const float* vp = vpos + ((long)n * CC + lane) * PS;
                if (n + 8 < NCAP) __builtin_prefetch(vp + 8 * CC * PS, 0, 0);
                float ln = s_cst[lane];
                #pragma unroll
                for (int q = 0; q < 4; ++q) {
                    const float4 vv = reinterpret_cast<const float4*>(vp)[q];
                    const int k = lane * CPAD + q * 4;
                    const float d0 = vv.x - s_mu[k + 0];
                    const float d1 = vv.y - s_mu[k + 1];
                    const float d2 = vv.z - s_mu[k + 2];
                    const float d3 = vv.w - s_mu[k + 3];
                    ln -= d0 * d0 * s_hinv[k + 0] + d1 * d1 * s_hinv[k + 1]
                        + d2 * d2 * s_hinv[k + 2] + d3 * d3 * s_hinv[k + 3];
                }
                s_r[n * RPAD + lane] = ln;
            }
            __syncthreads();
            // softmax over c, one thread per n
            for (int n = tid; n < NCAP; n += 256) {
                float mx = -3.4e38f;
                #pragma unroll
                for (int c = 0; c < CC; ++c) mx = fmaxf(mx, s_r[n * RPAD + c]);
                float s = 0.0f;
                #pragma unroll
                for (int c = 0; c < CC; ++c) {
                    const float e = __expf(s_r[n * RPAD + c] - mx);
                    s_r[n * RPAD + c] = e;
                    s += e;
                }
                const float inv = 1.0f / s;
                #pragma unroll
                for (int c = 0; c < CC; ++c) s_r[n * RPAD + c] *= inv;
            }
        }
        if (tid < CC) s_rsum[tid] = 0.0f;
        for (int idx = tid; idx < CC * PS; idx += 256) {
            const int c = idx >> 4, p = idx & 15;
            s_amu[c * CPAD + p] = 0.0f;
            s_am2[c * CPAD + p] = 0.0f;
        }
        __syncthreads();

        {   // r_sum[c] = sum_n r[n][c]
            const int c = tid & 31, g = tid >> 5;
            float part = 0.0f;
            for (int n = g; n < NCAP; n += 8) part += s_r[n * RPAD + c];
            atomicAdd(&s_rsum[c], part);
        }
        __syncthreads();

        // ---------------- M step: mu, sigma --------------------------------
        {
            float accmu[PS], accm2[PS];
            #pragma unroll
            for (int p = 0; p < PS; ++p) { accmu[p] = 0.0f; accm2[p] = 0.0f; }
            const float invrs = 1.0f / (s_rsum[lane] + eps);
            for (int n = wv; n < NCAP; n += 8) {
                const float* vp = vpos + ((long)n * CC + lane) * PS;
                if (n + 8 < NCAP) __builtin_prefetch(vp + 8 * CC * PS, 0, 0);
                const float coeff = s_r[n * RPAD + lane] * invrs;
                #pragma unroll
                for (int q = 0; q < 4; ++q) {
                    const float4 vv = reinterpret_cast<const float4*>(vp)[q];
                    accmu[q * 4 + 0] += coeff * vv.x; accm2[q * 4 + 0] += coeff * vv.x * vv.x;
                    accmu[q * 4 + 1] += coeff * vv.y; accm2[q * 4 + 1] += coeff * vv.y * vv.y;
                    accmu[q * 4 + 2] += coeff * vv.z; accm2[q * 4 + 2] += coeff * vv.z * vv.z;
                    accmu[q * 4 + 3] += coeff * vv.w; accm2[q * 4 + 3] += coeff * vv.w * vv.w;
                }
            }
            #pragma unroll
            for (int p = 0; p < PS; ++p) {
                atomicAdd(&s_amu[lane * CPAD + p], accmu[p]);
                atomicAdd(&s_am2[lane * CPAD + p], accm2[p]);
            }
        }
        __syncthreads();

        // sigma^2 = Sum c*v^2 - mu^2*(2 - Sum c) + eps   (exact, one pass)
        for (int idx = tid; idx < CC * PS; idx += 256) {
            const int c = idx >> 4, p = idx & 15, k = c * CPAD + p;
            const float rs   = s_rsum[c];
            const float csum = rs / (rs + eps);
            const float mu   = s_amu[k];
            const float sig  = s_am2[k] - mu * mu * (2.0f - csum) + eps;
            s_mu[k]   = mu;
            s_sig[k]  = sig;
            s_hinv[k] = 0.5f / sig;
        }
        __syncthreads();

        // activation + constants for next E step
        if (tid < CC) {
            const int c = tid;
            const float rs = s_rsum[c];
            float halflog = 0.0f;
            #pragma unroll
            for (int p = 0; p < PS; ++p) halflog += 0.5f * __logf(s_sig[c * CPAD + p]);
            const float cost = (16.0f * beta_u[c] + halflog) * rs;
            const float aa   = 1.0f / (1.0f + __expf(-lam * (beta_a[c] - cost)));
            s_aout[c] = aa;
            s_cst[c]  = __logf(aa) - 8.0f * LN2PI - halflog;
        }
        __syncthreads();
    }

    // ---------------- outputs ----------------------------------------------
    for (int idx = tid; idx < CC * PS; idx += 256) {
        const int c = idx >> 4, p = idx & 15;
        out[(long)pos * (CC * PS) + c * PS + p] = s_mu[c * CPAD + p];
    }
    if (tid < CC)
        out[(long)NPOS * CC * PS + (long)pos * CC + tid] = s_aout[tid];
}

// ---------------------------------------------------------------------------
extern "C" void kernel_launch(void* const* d_in, const int* in_sizes, int n_in,
                              void* d_out, int out_size, void* d_ws, size_t ws_size,
                              hipStream_t stream) {
    const float* x  = (const float*)d_in[0];   // (8,14,14,32,16)
    const float* a  = (const float*)d_in[1];   // (8,14,14,32,1)
    const float* w  = (const float*)d_in[2];   // (288,32,4,4)
    const float* bu = (const float*)d_in[3];   // (32,1)
    const float* ba = (const float*)d_in[4];   // (32,1)
    float* out = (float*)d_out;
    float* v   = (float*)d_ws;                 // 288*288*32*16 floats (~170 MB)

    // 288 n * 72 pos-tiles = 20736 waves; 8 waves/block -> 2592 blocks
    convcaps_votes_wmma<<<dim3(2592), dim3(256), 0, stream>>>(x, w, v);
    convcaps_em_routing<<<dim3(NPOS), dim3(256), 0, stream>>>(v, a, bu, ba, out);
}